// GINFeatures_13675175870656
// MI455X (gfx1250) — compile-verified
//
#include <hip/hip_runtime.h>
#include <hip/hip_bf16.h>

typedef __attribute__((ext_vector_type(2))) float v2f;
typedef __attribute__((ext_vector_type(8))) float v8f;

#define N_NODES  100000
#define N_EDGES  1600000
#define DIM      64
#define N_LAYERS 3
#define N_GRAPHS 256
#define BN_EPS   1e-5f

// ---------------------------------------------------------------- utilities
__global__ __launch_bounds__(256) void copy_f4_kernel(float4* __restrict__ dst,
                                                      const float4* __restrict__ src,
                                                      int n4) {
    int i = blockIdx.x * 256 + threadIdx.x;
    if (i < n4) dst[i] = src[i];
}

__global__ __launch_bounds__(256) void zero_kernel(float* __restrict__ p, int n) {
    int i = blockIdx.x * 256 + threadIdx.x;
    if (i < n) p[i] = 0.0f;
}

// ------------------------------------------------------- GIN edge scatter-add
// agg was pre-initialized with x; add sum_{j->i} x_j. One feature per lane:
// lanes of a wave hit 32 consecutive floats -> coalesced atomics.
__global__ __launch_bounds__(256) void edge_scatter_kernel(float* __restrict__ agg,
                                                           const float* __restrict__ x,
                                                           const int* __restrict__ src,
                                                           const int* __restrict__ dst) {
    long long tid = (long long)blockIdx.x * 256 + threadIdx.x;
    int e = (int)(tid >> 6);
    int f = (int)(tid & 63);
    if (e < N_EDGES) {
        int s = src[e];
        int d = dst[e];
        atomicAdd(&agg[(long long)d * DIM + f], x[(long long)s * DIM + f]);
    }
}

// ------------------------------------------------------------ WMMA GEMM no.1
// h2 = A @ W + bias.  Block: 8 waves -> 32 rows x 64 cols output tile.
// Uses V_WMMA_F32_16X16X4_F32 (full fp32, 16 steps over K=64).
__global__ __launch_bounds__(256) void gemm1_wmma_kernel(float* __restrict__ h2,
                                                         const float* __restrict__ A,
                                                         const float* __restrict__ W,
                                                         const float* __restrict__ bias) {
    __shared__ float sW[DIM * DIM];   // 16 KB
    __shared__ float sA[32 * DIM];    //  8 KB

    int tid = threadIdx.x;

    // Stage W (4096 f32 = 1024 float4, 4 per thread) and A tile (512 float4, 2/thread)
    const float4* W4 = (const float4*)W;
    float4* sW4 = (float4*)sW;
#pragma unroll
    for (int i = 0; i < 4; ++i) sW4[tid + 256 * i] = W4[tid + 256 * i];

    long long rowBase = (long long)blockIdx.x * 32;
    const float4* A4 = (const float4*)(A + rowBase * DIM);
    float4* sA4 = (float4*)sA;
    sA4[tid]       = A4[tid];
    sA4[tid + 256] = A4[tid + 256];
    __syncthreads();

    int wave = tid >> 5;
    int lane = tid & 31;
    int rowTile = (wave >> 2) * 16;   // 0 or 16
    int c0      = (wave & 3) * 16;    // 0,16,32,48
    int m    = lane & 15;
    int half = lane >> 4;

    v8f acc = {};
#pragma unroll
    for (int k0 = 0; k0 < DIM; k0 += 4) {
        int ka = k0 + 2 * half;
        v2f a, b;
        a.x = sA[(rowTile + m) * DIM + ka];
        a.y = sA[(rowTile + m) * DIM + ka + 1];
        b.x = sW[ka * DIM + c0 + m];
        b.y = sW[(ka + 1) * DIM + c0 + m];
        acc = __builtin_amdgcn_wmma_f32_16x16x4_f32(false, a, false, b,
                                                    (short)0, acc, false, false);
    }

    float bs = bias[c0 + m];
#pragma unroll
    for (int r = 0; r < 8; ++r) {
        long long row = rowBase + rowTile + r + 8 * half;
        h2[row * DIM + c0 + m] = acc[r] + bs;
    }
}

// ----------------------------------------------------- BN column sum / sumsq
__global__ __launch_bounds__(256) void bn_stats_kernel(float* __restrict__ stats,
                                                       const float* __restrict__ h) {
    int f  = threadIdx.x & 63;
    int rl = threadIdx.x >> 6;   // 0..3
    float s = 0.0f, q = 0.0f;
    for (long long r = (long long)blockIdx.x * 4 + rl; r < N_NODES;
         r += (long long)gridDim.x * 4) {
        float v = h[r * DIM + f];
        s += v;
        q += v * v;
    }
    __shared__ float ss[256], sq[256];
    ss[threadIdx.x] = s;
    sq[threadIdx.x] = q;
    __syncthreads();
    if (threadIdx.x < 64) {
        s = ss[f] + ss[f + 64] + ss[f + 128] + ss[f + 192];
        q = sq[f] + sq[f + 64] + sq[f + 128] + sq[f + 192];
        atomicAdd(&stats[f], s);
        atomicAdd(&stats[64 + f], q);
    }
}

// scale = gamma * rsqrt(var+eps); shift = beta - mu*scale
__global__ __launch_bounds__(64) void bn_finalize_kernel(float* __restrict__ scale,
                                                         float* __restrict__ shift,
                                                         const float* __restrict__ stats,
                                                         const float* __restrict__ gamma,
                                                         const float* __restrict__ beta) {
    int f = threadIdx.x;
    float inv_n = 1.0f / (float)N_NODES;
    float mu  = stats[f] * inv_n;
    float var = stats[64 + f] * inv_n - mu * mu;
    float inv = rsqrtf(var + BN_EPS);
    float g   = gamma[f] * inv;
    scale[f] = g;
    shift[f] = beta[f] - mu * g;
}

// ------------------------------------------------------------ WMMA GEMM no.2
// xout = relu( relu(h2*scale+shift) @ W + bias ).  BN+ReLU fused into A staging.
__global__ __launch_bounds__(256) void gemm2_wmma_kernel(float* __restrict__ xout,
                                                         const float* __restrict__ h2,
                                                         const float* __restrict__ W,
                                                         const float* __restrict__ bias,
                                                         const float* __restrict__ scale,
                                                         const float* __restrict__ shift) {
    __shared__ float sW[DIM * DIM];
    __shared__ float sA[32 * DIM];

    int tid = threadIdx.x;

    const float4* W4 = (const float4*)W;
    float4* sW4 = (float4*)sW;
#pragma unroll
    for (int i = 0; i < 4; ++i) sW4[tid + 256 * i] = W4[tid + 256 * i];

    long long rowBase = (long long)blockIdx.x * 32;
    const float4* H4 = (const float4*)(h2 + rowBase * DIM);
    const float4* sc4 = (const float4*)scale;
    const float4* sh4 = (const float4*)shift;
    float4* sA4 = (float4*)sA;
#pragma unroll
    for (int i = 0; i < 2; ++i) {
        int idx = tid + 256 * i;          // float4 index within 32x64 tile
        float4 v  = H4[idx];
        float4 sc = sc4[idx & 15];        // 16 float4 per row -> feature group
        float4 sh = sh4[idx & 15];
        v.x = fmaxf(fmaf(v.x, sc.x, sh.x), 0.0f);
        v.y = fmaxf(fmaf(v.y, sc.y, sh.y), 0.0f);
        v.z = fmaxf(fmaf(v.z, sc.z, sh.z), 0.0f);
        v.w = fmaxf(fmaf(v.w, sc.w, sh.w), 0.0f);
        sA4[idx] = v;
    }
    __syncthreads();

    int wave = tid >> 5;
    int lane = tid & 31;
    int rowTile = (wave >> 2) * 16;
    int c0      = (wave & 3) * 16;
    int m    = lane & 15;
    int half = lane >> 4;

    v8f acc = {};
#pragma unroll
    for (int k0 = 0; k0 < DIM; k0 += 4) {
        int ka = k0 + 2 * half;
        v2f a, b;
        a.x = sA[(rowTile + m) * DIM + ka];
        a.y = sA[(rowTile + m) * DIM + ka + 1];
        b.x = sW[ka * DIM + c0 + m];
        b.y = sW[(ka + 1) * DIM + c0 + m];
        acc = __builtin_amdgcn_wmma_f32_16x16x4_f32(false, a, false, b,
                                                    (short)0, acc, false, false);
    }

    float bs = bias[c0 + m];
#pragma unroll
    for (int r = 0; r < 8; ++r) {
        long long row = rowBase + rowTile + r + 8 * half;
        xout[row * DIM + c0 + m] = fmaxf(acc[r] + bs, 0.0f);
    }
}

// ------------------------------------------------------------- mean pooling
__global__ __launch_bounds__(256) void pool_scatter_kernel(float* __restrict__ sums,
                                                           float* __restrict__ counts,
                                                           const float* __restrict__ x,
                                                           const int* __restrict__ batch) {
    long long tid = (long long)blockIdx.x * 256 + threadIdx.x;
    int node = (int)(tid >> 6);
    int f    = (int)(tid & 63);
    if (node < N_NODES) {
        int g = batch[node];
        atomicAdd(&sums[(long long)g * DIM + f], x[(long long)node * DIM + f]);
        if (f == 0) atomicAdd(&counts[g], 1.0f);
    }
}

__global__ __launch_bounds__(256) void pool_finalize_kernel(float* __restrict__ out,
                                                            const float* __restrict__ sums,
                                                            const float* __restrict__ counts) {
    int tid = blockIdx.x * 256 + threadIdx.x;   // 0 .. 16383
    int g = tid >> 6;
    out[tid] = sums[tid] / fmaxf(counts[g], 1.0f);
}

// ----------------------------------------------------------------- launcher
extern "C" void kernel_launch(void* const* d_in, const int* in_sizes, int n_in,
                              void* d_out, int out_size, void* d_ws, size_t ws_size,
                              hipStream_t stream) {
    const float* x     = (const float*)d_in[0];
    const int*   ei    = (const int*)d_in[1];     // [2, N_EDGES]
    const int*   batch = (const int*)d_in[2];     // [N_NODES]
    const float* W1    = (const float*)d_in[3];   // [3,64,64]
    const float* b1    = (const float*)d_in[4];   // [3,64]
    const float* gamma = (const float*)d_in[5];
    const float* beta  = (const float*)d_in[6];
    const float* W2    = (const float*)d_in[7];
    const float* b2    = (const float*)d_in[8];

    const int* e_src = ei;
    const int* e_dst = ei + N_EDGES;

    const size_t FEAT_BYTES = (size_t)N_NODES * DIM * sizeof(float); // 25.6 MB
    char* ws = (char*)d_ws;
    float* agg   = (float*)(ws);
    float* h2    = (float*)(ws + FEAT_BYTES);
    float* xbuf  = (float*)(ws + 2 * FEAT_BYTES);
    float* stats = (float*)(ws + 3 * FEAT_BYTES);            // 128 f32
    float* scale = stats + 128;                              // 64 f32
    float* shift = scale + 64;                               // 64 f32
    float* sums  = shift + 64;                               // 256*64 f32
    float* cnts  = sums + N_GRAPHS * DIM;                    // 256 f32

    const int n_feat   = N_NODES * DIM;          // 6,400,000
    const int n_feat4  = n_feat / 4;             // 1,600,000
    const int copyGrid = n_feat4 / 256;          // 6250 (exact)
    const int gemmGrid = N_NODES / 32;           // 3125 (exact)
    const int scatGrid = (int)(((long long)N_EDGES * DIM) / 256);  // 400,000 (exact)
    const int poolGrid = n_feat / 256;           // 25,000 (exact)

    const float* cur = x;
    for (int l = 0; l < N_LAYERS; ++l) {
        // agg = cur ; agg[dst] += cur[src]
        copy_f4_kernel<<<copyGrid, 256, 0, stream>>>((float4*)agg, (const float4*)cur, n_feat4);
        edge_scatter_kernel<<<scatGrid, 256, 0, stream>>>(agg, cur, e_src, e_dst);

        // h2 = agg @ W1[l] + b1[l]
        gemm1_wmma_kernel<<<gemmGrid, 256, 0, stream>>>(h2, agg,
                                                        W1 + (size_t)l * DIM * DIM,
                                                        b1 + (size_t)l * DIM);

        // batch-norm statistics over rows
        zero_kernel<<<1, 128, 0, stream>>>(stats, 128);
        bn_stats_kernel<<<512, 256, 0, stream>>>(stats, h2);
        bn_finalize_kernel<<<1, 64, 0, stream>>>(scale, shift, stats,
                                                 gamma + (size_t)l * DIM,
                                                 beta + (size_t)l * DIM);

        // xbuf = relu( relu(BN(h2)) @ W2[l] + b2[l] )
        gemm2_wmma_kernel<<<gemmGrid, 256, 0, stream>>>(xbuf, h2,
                                                        W2 + (size_t)l * DIM * DIM,
                                                        b2 + (size_t)l * DIM,
                                                        scale, shift);
        cur = xbuf;
    }

    // global mean pool
    zero_kernel<<<(N_GRAPHS * DIM + N_GRAPHS + 255) / 256, 256, 0, stream>>>(
        sums, N_GRAPHS * DIM + N_GRAPHS);
    pool_scatter_kernel<<<poolGrid, 256, 0, stream>>>(sums, cnts, cur, batch);
    pool_finalize_kernel<<<(N_GRAPHS * DIM) / 256, 256, 0, stream>>>((float*)d_out, sums, cnts);
}